// GCN_51616916964127
// MI455X (gfx1250) — compile-verified
//
#include <hip/hip_runtime.h>
#include <math.h>

typedef __attribute__((ext_vector_type(2))) float v2f;
typedef __attribute__((ext_vector_type(8))) float v8f;

// ---------------------------------------------------------------------------
// Degree / normalization
// ---------------------------------------------------------------------------
__global__ void k_init_deg(int* deg, int N) {
    int i = blockIdx.x * blockDim.x + threadIdx.x;
    if (i < N) deg[i] = 1;  // self-loop
}

__global__ void k_count(const int* __restrict__ dst, int* deg, int E) {
    int e = blockIdx.x * blockDim.x + threadIdx.x;
    if (e < E) atomicAdd(&deg[dst[e]], 1);
}

// deg (int) and dinv (float) alias the same buffer: read int, write float.
__global__ void k_dinv(const int* deg, float* dinv, int N) {
    int i = blockIdx.x * blockDim.x + threadIdx.x;
    if (i < N) {
        float f = (float)deg[i];          // always >= 1
        dinv[i] = 1.0f / sqrtf(f);
    }
}

// ---------------------------------------------------------------------------
// Layer 1 transform: y1 = dinv * (x @ W1), via V_WMMA_F32_16X16X4_F32.
// A (16x4 f32): lanes 0-15 hold K=0,1 (the two real input features),
//               lanes 16-31 hold K=2,3 (zero pad).
// B (4x16 f32): lanes 0-15 hold K=0,1 for column N=lane; lanes 16-31 K=2,3 (pad).
// D (16x16 f32): VGPR v -> row M = v + 8*(lane/16), col N = lane%16.
// One wave handles 16 nodes; cols 0..3 of D are xw1.
// ---------------------------------------------------------------------------
__global__ void k_xw1(const float* __restrict__ x, const float* __restrict__ W1,
                      const float* __restrict__ dinv,
                      float* __restrict__ y1, float* __restrict__ agg1, int N) {
    int lane = threadIdx.x & 31;
    int wave = threadIdx.x >> 5;
    int base = (blockIdx.x * (blockDim.x >> 5) + wave) * 16;
    if (base >= N) return;                 // wave-uniform exit
    int m    = lane & 15;
    int half = lane >> 4;
    int node = base + m;
    bool full = (base + 16 <= N);          // wave-uniform

    v2f a = {0.0f, 0.0f};
    if (half == 0 && node < N) {           // K=0,1 real; K=2,3 zero
        a.x = x[node * 2 + 0];
        a.y = x[node * 2 + 1];
    }
    v2f b = {0.0f, 0.0f};
    if (half == 0 && m < 4) {              // W1 is [2][4] row-major
        b.x = W1[0 * 4 + m];
        b.y = W1[1 * 4 + m];
    }
    v8f c = {};
    v8f d = __builtin_amdgcn_wmma_f32_16x16x4_f32(
        false, a, false, b, (short)0, c, false, false);

    int nb = base + 8 * half;              // 8 consecutive node ids per half
    if (full) {
        if (m < 4) {
            float dv[8];
            #pragma unroll
            for (int v = 0; v < 8; ++v) dv[v] = dinv[nb + v];  // contiguous -> b128
            #pragma unroll
            for (int v = 0; v < 8; ++v) {
                float val = dv[v] * d[v];      // y = dinv * xw
                y1[(nb + v) * 4 + m]   = val;
                agg1[(nb + v) * 4 + m] = val;  // self-loop init: acc = y[d]
            }
        }
    } else {
        if (m < 4) {
            #pragma unroll
            for (int v = 0; v < 8; ++v) {
                int nd = nb + v;
                if (nd < N) {
                    float val = dinv[nd] * d[v];
                    y1[nd * 4 + m]   = val;
                    agg1[nd * 4 + m] = val;
                }
            }
        }
    }
}

// ---------------------------------------------------------------------------
// Edge scatter, layer 1: agg1[dst] += y1[src]  (4 floats)
// ---------------------------------------------------------------------------
__global__ void k_scatter1(const int* __restrict__ src, const int* __restrict__ dst,
                           const float* __restrict__ y1, float* agg1, int E) {
    int e = blockIdx.x * blockDim.x + threadIdx.x;
    if (e >= E) return;
    int s = src[e];
    int t = dst[e];
    float4 y = *(const float4*)(y1 + (size_t)s * 4);
    float* p = agg1 + (size_t)t * 4;
    atomicAdd(p + 0, y.x);
    atomicAdd(p + 1, y.y);
    atomicAdd(p + 2, y.z);
    atomicAdd(p + 3, y.w);
}

// ---------------------------------------------------------------------------
// Layer 1 finalize + layer 2 transform:
// h1 = tanh(dinv*agg1 + b1); y2 = dinv * (h1 @ W2), via WMMA f32 16x16x4.
// K=4 exactly: lane<16 computes h1[node][0..1], lane>=16 h1[node][2..3].
// ---------------------------------------------------------------------------
__global__ void k_layer2(const float* __restrict__ agg1, const float* __restrict__ dinv,
                         const float* __restrict__ b1, const float* __restrict__ W2,
                         float* __restrict__ y2, float* __restrict__ agg2, int N) {
    int lane = threadIdx.x & 31;
    int wave = threadIdx.x >> 5;
    int base = (blockIdx.x * (blockDim.x >> 5) + wave) * 16;
    if (base >= N) return;
    int m    = lane & 15;
    int half = lane >> 4;
    int node = base + m;
    int kb   = half * 2;                   // K offset handled by this lane
    bool full = (base + 16 <= N);          // wave-uniform

    v2f a = {0.0f, 0.0f};
    if (node < N) {
        float dv = dinv[node];
        a.x = tanhf(dv * agg1[node * 4 + kb + 0] + b1[kb + 0]);
        a.y = tanhf(dv * agg1[node * 4 + kb + 1] + b1[kb + 1]);
    }
    v2f b = {0.0f, 0.0f};
    if (m < 2) {                           // W2 is [4][2] row-major; cols 0..1
        b.x = W2[(kb + 0) * 2 + m];
        b.y = W2[(kb + 1) * 2 + m];
    }
    v8f c = {};
    v8f d = __builtin_amdgcn_wmma_f32_16x16x4_f32(
        false, a, false, b, (short)0, c, false, false);

    int nb = base + 8 * half;
    if (full) {
        if (m < 2) {
            float dv[8];
            #pragma unroll
            for (int v = 0; v < 8; ++v) dv[v] = dinv[nb + v];  // contiguous -> b128
            #pragma unroll
            for (int v = 0; v < 8; ++v) {
                float val = dv[v] * d[v];
                y2[(nb + v) * 2 + m]   = val;
                agg2[(nb + v) * 2 + m] = val;  // self-loop init
            }
        }
    } else {
        if (m < 2) {
            #pragma unroll
            for (int v = 0; v < 8; ++v) {
                int nd = nb + v;
                if (nd < N) {
                    float val = dinv[nd] * d[v];
                    y2[nd * 2 + m]   = val;
                    agg2[nd * 2 + m] = val;
                }
            }
        }
    }
}

// ---------------------------------------------------------------------------
// Edge scatter, layer 2: agg2[dst] += y2[src]  (2 floats)
// ---------------------------------------------------------------------------
__global__ void k_scatter2(const int* __restrict__ src, const int* __restrict__ dst,
                           const float* __restrict__ y2, float* agg2, int E) {
    int e = blockIdx.x * blockDim.x + threadIdx.x;
    if (e >= E) return;
    int s = src[e];
    int t = dst[e];
    float2 y = *(const float2*)(y2 + (size_t)s * 2);
    float* p = agg2 + (size_t)t * 2;
    atomicAdd(p + 0, y.x);
    atomicAdd(p + 1, y.y);
}

// ---------------------------------------------------------------------------
// Layer 2 finalize + fused 2x2 classifier: out = tanh(dinv*agg2 + b2) @ Wc + bc
// ---------------------------------------------------------------------------
__global__ void k_final(const float* __restrict__ agg2, const float* __restrict__ dinv,
                        const float* __restrict__ b2, const float* __restrict__ Wc,
                        const float* __restrict__ bc, float* __restrict__ out, int N) {
    int i = blockIdx.x * blockDim.x + threadIdx.x;
    if (i >= N) return;
    float dv = dinv[i];
    float h0 = tanhf(dv * agg2[i * 2 + 0] + b2[0]);
    float h1 = tanhf(dv * agg2[i * 2 + 1] + b2[1]);
    float2 o;
    o.x = h0 * Wc[0] + h1 * Wc[2] + bc[0];   // Wc row-major [2][2]
    o.y = h0 * Wc[1] + h1 * Wc[3] + bc[1];
    *(float2*)(out + (size_t)i * 2) = o;
}

// ---------------------------------------------------------------------------
extern "C" void kernel_launch(void* const* d_in, const int* in_sizes, int n_in,
                              void* d_out, int out_size, void* d_ws, size_t ws_size,
                              hipStream_t stream) {
    const float* x  = (const float*)d_in[0];
    const int*   ei = (const int*)d_in[1];
    const float* W1 = (const float*)d_in[2];
    const float* b1 = (const float*)d_in[3];
    const float* W2 = (const float*)d_in[4];
    const float* b2 = (const float*)d_in[5];
    const float* Wc = (const float*)d_in[6];
    const float* bc = (const float*)d_in[7];
    float* out = (float*)d_out;

    const int N = in_sizes[0] / 2;   // IN = 2
    const int E = in_sizes[1] / 2;   // edge_index is [2, E]
    const int* src = ei;
    const int* dst = ei + E;

    // Workspace layout (floats): [0,N) deg(int)/dinv | [N,5N) y1 | [5N,9N) agg1
    //                            [9N,11N) y2 | [11N,13N) agg2   -> 26 MB total
    float* ws   = (float*)d_ws;
    int*   deg  = (int*)ws;
    float* dinv = ws;
    float* y1   = ws + (size_t)N;
    float* agg1 = ws + (size_t)5 * N;
    float* y2   = ws + (size_t)9 * N;
    float* agg2 = ws + (size_t)11 * N;

    const int T = 256;                       // 8 wave32 per block
    dim3 gN((N + T - 1) / T);
    dim3 gE((E + T - 1) / T);
    const int nodesPerBlock = (T / 32) * 16; // 16 nodes per wave
    dim3 gW((N + nodesPerBlock - 1) / nodesPerBlock);

    k_init_deg<<<gN, T, 0, stream>>>(deg, N);
    k_count   <<<gE, T, 0, stream>>>(dst, deg, E);
    k_dinv    <<<gN, T, 0, stream>>>(deg, dinv, N);
    k_xw1     <<<gW, T, 0, stream>>>(x, W1, dinv, y1, agg1, N);
    k_scatter1<<<gE, T, 0, stream>>>(src, dst, y1, agg1, E);
    k_layer2  <<<gW, T, 0, stream>>>(agg1, dinv, b1, W2, y2, agg2, N);
    k_scatter2<<<gE, T, 0, stream>>>(src, dst, y2, agg2, E);
    k_final   <<<gN, T, 0, stream>>>(agg2, dinv, b2, Wc, bc, out, N);
}